// ProductSpaceGNN_19937238188301
// MI455X (gfx1250) — compile-verified
//
#include <hip/hip_runtime.h>

typedef __attribute__((ext_vector_type(2))) float v2f;
typedef __attribute__((ext_vector_type(8))) float v8f;

static constexpr int   kNodes = 50000;
static constexpr int   kEdges = 800000;
static constexpr float kLnEps = 1e-5f;

// ---------------------------------------------------------------------------
// Degree count (atomic) and 1/max(deg,1)
// ---------------------------------------------------------------------------
__global__ void deg_count_kernel(const int* __restrict__ tgt,
                                 float* __restrict__ deg, int nEdges) {
  int i = blockIdx.x * blockDim.x + threadIdx.x;
  if (i < nEdges) atomicAdd(&deg[tgt[i]], 1.0f);
}

__global__ void inv_deg_kernel(float* __restrict__ deg, int n) {
  int i = blockIdx.x * blockDim.x + threadIdx.x;
  if (i < n) deg[i] = 1.0f / fmaxf(deg[i], 1.0f);
}

// ---------------------------------------------------------------------------
// Edge scatter: agg[tgt] += h[src]   (DIN/4 lanes per edge, float4 gather)
// ---------------------------------------------------------------------------
template <int DIN>
__global__ void scatter_sum_kernel(const float* __restrict__ h,
                                   const int* __restrict__ src,
                                   const int* __restrict__ tgt,
                                   float* __restrict__ agg, int nEdges) {
  constexpr int TPE = DIN / 4;  // threads per edge
  unsigned idx = blockIdx.x * blockDim.x + threadIdx.x;
  unsigned e = idx / TPE;
  if (e >= (unsigned)nEdges) return;
  int c = (int)(idx % TPE) * 4;
  int s = src[e];
  int t = tgt[e];
  const float4 v = *reinterpret_cast<const float4*>(h + (size_t)s * DIN + c);
  float* dst = agg + (size_t)t * DIN + c;
  atomicAdd(dst + 0, v.x);
  atomicAdd(dst + 1, v.y);
  atomicAdd(dst + 2, v.z);
  atomicAdd(dst + 3, v.w);
}

// ---------------------------------------------------------------------------
// Fused SAGE layer: out = LN( (agg*inv_deg)@Wl^T + h@Wr^T + b ) [; ReLU | L2]
// One block = 16 node rows x DOUT cols.  One wave per 16-wide column tile.
// fp32 WMMA 16x16x4: A lane<16 -> K{0,1}, lane>=16 -> K{2,3} (float2/lane);
// B[k][n] = W[n][k] -> same float2 pattern on row-major W[DOUT][DIN].
// C VGPR r: lanes 0-15 -> M=r, lanes 16-31 -> M=r+8.
// ---------------------------------------------------------------------------
template <int DIN, int DOUT, bool RELU, bool L2NORM>
__global__ __launch_bounds__(32 * (DOUT / 16))
void sage_gemm_ln_kernel(const float* __restrict__ agg,
                         const float* __restrict__ h,
                         const float* __restrict__ inv_deg,
                         const float* __restrict__ Wl,
                         const float* __restrict__ Wr,
                         const float* __restrict__ bias,
                         const float* __restrict__ gamma,
                         const float* __restrict__ beta,
                         float* __restrict__ out) {
  constexpr int NT = 32 * (DOUT / 16);
  __shared__ float tile[16][DOUT + 1];  // +1 pad: conflict-free row scans
  __shared__ float s_mu[16];
  __shared__ float s_rs[16];

  const int tid = threadIdx.x;
  const int lane = tid & 31;
  const int wave = tid >> 5;
  const int row_base = blockIdx.x * 16;
  const int n_base = wave * 16;

  const int m = lane & 15;               // A row within tile / B col within tile
  const int khalf = (lane >> 4) << 1;    // 0 or 2: which K-pair this lane holds
  const int row = row_base + m;
  const float scale = inv_deg[row];

  const float* aggp = agg + (size_t)row * DIN + khalf;
  const float* hp   = h   + (size_t)row * DIN + khalf;
  const float* wlp  = Wl  + (size_t)(n_base + m) * DIN + khalf;
  const float* wrp  = Wr  + (size_t)(n_base + m) * DIN + khalf;

  v8f c = {0.f, 0.f, 0.f, 0.f, 0.f, 0.f, 0.f, 0.f};

#pragma unroll 4
  for (int k = 0; k < DIN; k += 4) {
    v2f a1 = *reinterpret_cast<const v2f*>(aggp + k);
    a1 = a1 * scale;  // fold mean-aggregation scale into A rows
    v2f b1 = *reinterpret_cast<const v2f*>(wlp + k);
    c = __builtin_amdgcn_wmma_f32_16x16x4_f32(false, a1, false, b1,
                                              (short)0, c, false, false);
    v2f a2 = *reinterpret_cast<const v2f*>(hp + k);
    v2f b2 = *reinterpret_cast<const v2f*>(wrp + k);
    c = __builtin_amdgcn_wmma_f32_16x16x4_f32(false, a2, false, b2,
                                              (short)0, c, false, false);
  }

  // Spill C tile (+bias) to LDS for the cross-feature LayerNorm.
  const int nIdx = n_base + (lane & 15);
  const float bn = bias[nIdx];
  const int rowOff = (lane >> 4) << 3;  // 0 or 8
#pragma unroll
  for (int r = 0; r < 8; ++r) tile[r + rowOff][nIdx] = c[r] + bn;
  __syncthreads();

  // Per-row mean / rstd (16 rows, one thread each; DOUT-long scans hit
  // distinct banks thanks to the +1 pad).
  if (tid < 16) {
    float s = 0.f;
#pragma unroll
    for (int j = 0; j < DOUT; ++j) s += tile[tid][j];
    const float mu = s * (1.0f / DOUT);
    float v = 0.f;
#pragma unroll
    for (int j = 0; j < DOUT; ++j) {
      const float d = tile[tid][j] - mu;
      v += d * d;
    }
    s_mu[tid] = mu;
    s_rs[tid] = rsqrtf(v * (1.0f / DOUT) + kLnEps);
  }
  __syncthreads();

  constexpr int TOTAL = 16 * DOUT;
  if (!L2NORM) {
#pragma unroll
    for (int i = tid; i < TOTAL; i += NT) {
      const int rr = i / DOUT, cc = i % DOUT;
      float val = (tile[rr][cc] - s_mu[rr]) * s_rs[rr] * gamma[cc] + beta[cc];
      if (RELU) val = fmaxf(val, 0.f);
      out[(size_t)(row_base + rr) * DOUT + cc] = val;
    }
  } else {
    // Apply LN in-place, then row-wise L2 normalize (final layer).
#pragma unroll
    for (int i = tid; i < TOTAL; i += NT) {
      const int rr = i / DOUT, cc = i % DOUT;
      tile[rr][cc] = (tile[rr][cc] - s_mu[rr]) * s_rs[rr] * gamma[cc] + beta[cc];
    }
    __syncthreads();
    if (tid < 16) {
      float ss = 0.f;
#pragma unroll
      for (int j = 0; j < DOUT; ++j) ss += tile[tid][j] * tile[tid][j];
      s_rs[tid] = 1.0f / fmaxf(sqrtf(ss), 1e-12f);
    }
    __syncthreads();
#pragma unroll
    for (int i = tid; i < TOTAL; i += NT) {
      const int rr = i / DOUT, cc = i % DOUT;
      out[(size_t)(row_base + rr) * DOUT + cc] = tile[rr][cc] * s_rs[rr];
    }
  }
}

// ---------------------------------------------------------------------------
// Launch
// ---------------------------------------------------------------------------
extern "C" void kernel_launch(void* const* d_in, const int* in_sizes, int n_in,
                              void* d_out, int out_size, void* d_ws,
                              size_t ws_size, hipStream_t stream) {
  const float* x = (const float*)d_in[0];
  const int* edge = (const int*)d_in[1];
  const int* src = edge;            // edge_index[0]
  const int* tgt = edge + kEdges;   // edge_index[1]

  const float* P[15];
  for (int i = 0; i < 15; ++i) P[i] = (const float*)d_in[2 + i];
  // P[0..4]  = Wl0, Wr0, b0, g0, be0   (128x128 layer)
  // P[5..9]  = Wl1, Wr1, b1, g1, be1   (128x128 layer)
  // P[10..14]= Wl2, Wr2, b2, g2, be2   (64x128 layer)

  float* ws = (float*)d_ws;
  float* deg = ws;  // [50000] -> becomes inv_deg in place
  size_t off = 50048;  // 64-float aligned
  float* agg = ws + off; off += (size_t)kNodes * 128;
  float* hA  = ws + off; off += (size_t)kNodes * 128;
  float* hB  = ws + off;

  const size_t aggBytes = (size_t)kNodes * 128 * sizeof(float);
  const int scatterBlocks = (kEdges * 32 + 255) / 256;
  const int rowBlocks = kNodes / 16;  // 3125, exact

  // Degrees (shared by all layers)
  hipMemsetAsync(deg, 0, kNodes * sizeof(float), stream);
  deg_count_kernel<<<(kEdges + 255) / 256, 256, 0, stream>>>(tgt, deg, kEdges);
  inv_deg_kernel<<<(kNodes + 255) / 256, 256, 0, stream>>>(deg, kNodes);

  // Layer 0: x -> hA
  hipMemsetAsync(agg, 0, aggBytes, stream);
  scatter_sum_kernel<128><<<scatterBlocks, 256, 0, stream>>>(x, src, tgt, agg, kEdges);
  sage_gemm_ln_kernel<128, 128, true, false><<<rowBlocks, 256, 0, stream>>>(
      agg, x, deg, P[0], P[1], P[2], P[3], P[4], hA);

  // Layer 1: hA -> hB
  hipMemsetAsync(agg, 0, aggBytes, stream);
  scatter_sum_kernel<128><<<scatterBlocks, 256, 0, stream>>>(hA, src, tgt, agg, kEdges);
  sage_gemm_ln_kernel<128, 128, true, false><<<rowBlocks, 256, 0, stream>>>(
      agg, hA, deg, P[5], P[6], P[7], P[8], P[9], hB);

  // Layer 2: hB -> out (no ReLU, row L2-normalize fused)
  hipMemsetAsync(agg, 0, aggBytes, stream);
  scatter_sum_kernel<128><<<scatterBlocks, 256, 0, stream>>>(hB, src, tgt, agg, kEdges);
  sage_gemm_ln_kernel<128, 64, false, true><<<rowBlocks, 128, 0, stream>>>(
      agg, hB, deg, P[10], P[11], P[12], P[13], P[14], (float*)d_out);
}